// Net_17446157156501
// MI455X (gfx1250) — compile-verified
//
#include <hip/hip_runtime.h>

typedef float v2f __attribute__((ext_vector_type(2)));
typedef float v4f __attribute__((ext_vector_type(4)));
typedef float v8f __attribute__((ext_vector_type(8)));

#define BETA   0.95f
#define THRESH 1.0f
#define NSTEPS 25
#define BATCH  8192
#define INF    784
#define HID    1000
#define OUTF   10

// ---------------------------------------------------------------------------
// Kernel 1: cur1[B][H] = x[B][IN] @ W1[H][IN]^T + b1   (f32 WMMA 16x16x4)
// One wave computes a 16x64 output tile (1 M-tile x 4 N-tiles), reusing the
// A fragment across 4 WMMAs per K-group. Out-of-range N columns are handled
// by CLAMPING the B row (garbage only lands in D columns that are never
// stored), so the inner loop has no divergence and no branches.
// ---------------------------------------------------------------------------
__global__ __launch_bounds__(256) void snn_fc1(const float* __restrict__ x,
                                               const float* __restrict__ W1,
                                               const float* __restrict__ b1,
                                               float* __restrict__ cur1) {
    const int tid   = threadIdx.x;
    const int lane  = tid & 31;
    const int wave  = tid >> 5;
    const int job   = blockIdx.x * 8 + wave;   // 0..8191
    const int mt    = job >> 4;                // 0..511  (16 rows each)
    const int ntg   = job & 15;                // 0..15   (64 cols each)
    const int m_in  = lane & 15;
    const int khalf = lane >> 4;               // 0 or 1

    const int arow = mt * 16 + m_in;
    const float* aptr = x + (size_t)arow * INF + khalf * 2;

    const int colb = ntg * 64 + m_in;
    // Clamp OOB columns to a valid row: their D columns are never stored.
    const float* bptr[4];
#pragma unroll
    for (int nt = 0; nt < 4; ++nt) {
        int col = colb + nt * 16;
        int crow = (col < HID) ? col : (HID - 1);
        bptr[nt] = W1 + (size_t)crow * INF + khalf * 2;
    }

    v8f acc[4];
#pragma unroll
    for (int nt = 0; nt < 4; ++nt) acc[nt] = (v8f){0,0,0,0,0,0,0,0};

    for (int g = 0; g < INF / 4; ++g) {
        v2f a = *(const v2f*)(aptr + g * 4);
#pragma unroll
        for (int nt = 0; nt < 4; ++nt) {
            v2f b = *(const v2f*)(bptr[nt] + g * 4);
            acc[nt] = __builtin_amdgcn_wmma_f32_16x16x4_f32(
                false, a, false, b, (short)0, acc[nt], false, false);
        }
    }

#pragma unroll
    for (int nt = 0; nt < 4; ++nt) {
        const int col = colb + nt * 16;
        if (col < HID) {
            const float bias = b1[col];
#pragma unroll
            for (int r = 0; r < 8; ++r) {
                const int orow = mt * 16 + r + khalf * 8;
                cur1[(size_t)orow * HID + col] = acc[nt][r] + bias;
            }
        }
    }
}

// ---------------------------------------------------------------------------
// Kernel 2: 25 LIF timesteps. Each block owns 32 batch rows end-to-end
// (rows are independent across steps -> no global sync needed).
//   Phase A: LIF layer-1 (mem1 via L2-resident workspace, float4 traffic),
//            spikes packed as bitmasks in LDS
//   Phase B: cur2 += spk1 @ W2^T via f32 WMMA (N padded 10->16 by clamping
//            the B row -- dead D columns absorb the garbage),
//            K=1000 split over 4 waves/M-tile, reduced with ds_add_f32
//   Phase C: LIF layer-2, write spk2_rec / mem2_rec
// ---------------------------------------------------------------------------
__global__ __launch_bounds__(256) void snn_steps(const float* __restrict__ cur1,
                                                 float* __restrict__ mem1,
                                                 const float* __restrict__ W2,
                                                 const float* __restrict__ b2,
                                                 float* __restrict__ out) {
    __shared__ unsigned spk_bits[32 * 32];      //  4 KB : 32 rows x 1024 bits
    __shared__ float    W2lds[OUTF * HID];      // 40 KB : W2 staged once
    __shared__ float    b2lds[16];
    __shared__ float    cur2lds[32 * 16];       //  2 KB : padded N=16
    __shared__ float    mem2lds[32 * OUTF];     //  mem2 state, block-resident

    const int tid  = threadIdx.x;
    const int lane = tid & 31;
    const int wave = tid >> 5;
    const int row0 = blockIdx.x * 32;

    for (int i = tid; i < OUTF * HID; i += 256) W2lds[i] = W2[i];
    if (tid < 16) b2lds[tid] = (tid < OUTF) ? b2[tid] : 0.0f;

    const float* cur1b = cur1 + (size_t)row0 * HID;
    float*       mem1b = mem1 + (size_t)row0 * HID;

    // Per-wave constants for Phase B (precomputed, loop-invariant).
    const int   mt_b    = wave >> 2;            // M-tile 0..1
    const int   ks_b    = wave & 3;             // K sub-partition
    const int   m_in    = lane & 15;
    const int   khalf   = lane >> 4;
    const int   arow_b  = mt_b * 16 + m_in;
    const float* w2ptr  = W2lds + (size_t)((m_in < OUTF) ? m_in : (OUTF - 1)) * HID
                        + khalf * 2;
    const unsigned* spkrow = spk_bits + arow_b * 32;

    for (int t = 0; t < NSTEPS; ++t) {
        __syncthreads();   // prev Phase C (and initial staging) complete

        // ---- Phase A: LIF-1, pack spikes as bits (float4 traffic) --------
#pragma unroll
        for (int i = 0; i < 4; ++i) {
            const int word_id = tid + 256 * i;      // 0..1023
            const int r    = word_id >> 5;          // row in tile (0..31)
            const int hb   = (word_id & 31) * 32;   // first h of this word
            const int base = r * HID + hb;
            const int nvec = (((HID - hb) < 32 ? (HID - hb) : 32)) >> 2; // 8 or 2
            unsigned bits = 0u;
            for (int q = 0; q < nvec; ++q) {
                v4f cur = *(const v4f*)(cur1b + base + q * 4);
                v4f mp  = (v4f){0,0,0,0};
                if (t != 0) mp = *(const v4f*)(mem1b + base + q * 4);
                v4f m;
#pragma unroll
                for (int e = 0; e < 4; ++e) {
                    const float mm = BETA * mp[e] + cur[e]
                                   - ((mp[e] > THRESH) ? THRESH : 0.0f);
                    m[e] = mm;
                    if (mm > THRESH) bits |= (1u << (q * 4 + e));
                }
                *(v4f*)(mem1b + base + q * 4) = m;
            }
            spk_bits[word_id] = bits;
        }
        // init cur2 accumulator with bias (512 slots, 2 per thread)
        {
            int s = tid;
            cur2lds[s] = ((s & 15) < OUTF) ? b2lds[s & 15] : 0.0f;
            s = tid + 256;
            cur2lds[s] = ((s & 15) < OUTF) ? b2lds[s & 15] : 0.0f;
        }
        __syncthreads();

        // ---- Phase B: spk1 @ W2^T via V_WMMA_F32_16X16X4_F32 -------------
        {
            v8f c = (v8f){0,0,0,0,0,0,0,0};
            for (int g = ks_b; g < HID / 4; g += 4) {
                const int k0 = g * 4 + khalf * 2;
                const unsigned word = spkrow[k0 >> 5];
                const int sh = k0 & 31;             // even -> sh+1 same word
                v2f a;
                a.x = ((word >> sh) & 1u) ? 1.0f : 0.0f;
                a.y = ((word >> (sh + 1)) & 1u) ? 1.0f : 0.0f;
                v2f b = *(const v2f*)(w2ptr + g * 4);   // clamped row, no branch
                c = __builtin_amdgcn_wmma_f32_16x16x4_f32(
                    false, a, false, b, (short)0, c, false, false);
            }
#pragma unroll
            for (int r = 0; r < 8; ++r) {
                const int orow = mt_b * 16 + r + khalf * 8;
                atomicAdd(&cur2lds[orow * 16 + m_in], c[r]);  // ds_add_f32
            }
        }
        __syncthreads();

        // ---- Phase C: LIF-2, record outputs ------------------------------
        for (int p = tid; p < 32 * OUTF; p += 256) {
            const int r = p / OUTF;
            const int o = p - r * OUTF;
            const float cur2 = cur2lds[r * 16 + o];
            const float m2p  = (t == 0) ? 0.0f : mem2lds[p];
            const float m2   = BETA * m2p + cur2
                             - ((m2p > THRESH) ? THRESH : 0.0f);
            mem2lds[p] = m2;
            const float spk2 = (m2 > THRESH) ? 1.0f : 0.0f;
            const size_t oidx = (size_t)t * (BATCH * OUTF)
                              + (size_t)(row0 + r) * OUTF + o;
            out[oidx] = spk2;                                   // spk2_rec
            out[(size_t)NSTEPS * BATCH * OUTF + oidx] = m2;     // mem2_rec
        }
    }
}

// ---------------------------------------------------------------------------
extern "C" void kernel_launch(void* const* d_in, const int* in_sizes, int n_in,
                              void* d_out, int out_size, void* d_ws, size_t ws_size,
                              hipStream_t stream) {
    const float* x  = (const float*)d_in[0];
    const float* W1 = (const float*)d_in[1];
    const float* b1 = (const float*)d_in[2];
    const float* W2 = (const float*)d_in[3];
    const float* b2 = (const float*)d_in[4];
    float* out  = (float*)d_out;
    float* cur1 = (float*)d_ws;                       // [8192 x 1000]
    float* mem1 = cur1 + (size_t)BATCH * HID;         // [8192 x 1000]

    snn_fc1<<<1024, 256, 0, stream>>>(x, W1, b1, cur1);
    snn_steps<<<BATCH / 32, 256, 0, stream>>>(cur1, mem1, W2, b2, out);
}